// FocalLoss_47021301956975
// MI455X (gfx1250) — compile-verified
//
#include <hip/hip_runtime.h>
#include <stdint.h>

typedef __attribute__((ext_vector_type(2))) float v2f;
typedef __attribute__((ext_vector_type(8))) float v8f;
typedef __attribute__((ext_vector_type(4))) unsigned int u32x4;
typedef __attribute__((ext_vector_type(8))) int i32x8;
typedef __attribute__((ext_vector_type(4))) int i32x4;

#define ALPHA_C 0.25f
#define MAXM 128

#ifndef __has_builtin
#define __has_builtin(x) 0
#endif

#if __has_builtin(__builtin_amdgcn_tensor_load_to_lds) && __has_builtin(__builtin_amdgcn_s_wait_tensorcnt)
#define USE_TDM 1
#else
#define USE_TDM 0
#endif

// ---------------- K0: zero the per-call flag buffers ----------------
__global__ void zero_ws_kernel(uint8_t* __restrict__ pos_tag,
                               unsigned long long* __restrict__ tag_best,
                               int nbytes, int ntag) {
    int i = blockIdx.x * blockDim.x + threadIdx.x;
    if (i < nbytes) pos_tag[i] = 0;
    if (i < ntag)   tag_best[i] = 0ull;
}

// ---------------- K1: IoU, per-anchor argmax, per-GT scatter-max ----------------
__global__ void iou_kernel(const float* __restrict__ anchors,   // [A,4]
                           const float* __restrict__ annot,     // [B,M,5]
                           float* __restrict__ iou_max_out,     // [B,A]
                           int*   __restrict__ iou_arg_out,     // [B,A]
                           unsigned long long* __restrict__ tag_best, // [B,M]
                           int A, int M) {
    const int b   = blockIdx.y;
    const int a   = blockIdx.x * blockDim.x + threadIdx.x;
    const int tid = threadIdx.x;

    __shared__ float sann[MAXM * 5];
    __shared__ unsigned long long stag[MAXM];

#if USE_TDM
    // Stage annot[b] (M*5 dwords) into LDS with the Tensor Data Mover.
    // One wave issues the DMA (TDM ignores EXEC; 1 instruction per wave),
    // waits on TENSORcnt, then __syncthreads() publishes to the block.
    if (tid < 32) {
        const unsigned n    = (unsigned)(M * 5);          // dwords in the tile
        const unsigned lds0 = (unsigned)(uintptr_t)(void*)sann;   // flat low32 == LDS offset
        const unsigned long long ga =
            (unsigned long long)(uintptr_t)(annot + (size_t)b * M * 5);
        u32x4 g0;
        g0[0] = 1u;                                        // count=1, is_restore=0
        g0[1] = lds0;                                      // lds_addr (bytes)
        g0[2] = (unsigned)(ga & 0xFFFFFFFFull);            // global_addr[31:0]
        g0[3] = (unsigned)((ga >> 32) & 0x01FFFFFFull)     // global_addr[56:32]
              | (2u << 30);                                // type = 2 ("image")
        i32x8 g1;
        g1[0] = (int)(2u << 16);                           // data_size = 2 -> 4 bytes
        g1[1] = (int)((n & 0xFFFFu) << 16);                // tensor_dim0[15:0]
        g1[2] = (int)(((n >> 16) & 0xFFFFu) | (1u << 16)); // tensor_dim0[31:16] | tensor_dim1=1
        g1[3] = (int)((n & 0xFFFFu) << 16);                // tensor_dim1 hi | tile_dim0 = n
        g1[4] = 1;                                         // tile_dim1 = 1, tile_dim2 = 0
        g1[5] = (int)n;                                    // tensor_dim0_stride[31:0]
        g1[6] = 0;                                         // stride hi / dim1_stride lo
        g1[7] = 0;
        i32x4 z4 = {0, 0, 0, 0};
        i32x8 z8 = {0, 0, 0, 0, 0, 0, 0, 0};
        __builtin_amdgcn_tensor_load_to_lds(g0, g1, z4, z4, z8, 0);
        __builtin_amdgcn_s_wait_tensorcnt(0);
    }
    for (int i = tid; i < M; i += blockDim.x) stag[i] = 0ull;
#else
    for (int i = tid; i < M * 5; i += blockDim.x) sann[i] = annot[(size_t)b * M * 5 + i];
    for (int i = tid; i < M;     i += blockDim.x) stag[i] = 0ull;
#endif
    __syncthreads();

    if (a < A) {
        const float ax0 = anchors[a * 4 + 0], ay0 = anchors[a * 4 + 1];
        const float ax1 = anchors[a * 4 + 2], ay1 = anchors[a * 4 + 3];
        const float aarea = (ax1 - ax0) * (ay1 - ay0);
        float best = -INFINITY; int bm = 0;
        for (int m = 0; m < M; ++m) {
            const float* g = &sann[m * 5];
            float iou;
            if (g[4] != -1.0f) {
                float barea = (g[2] - g[0]) * (g[3] - g[1]);
                float iw = fmaxf(fminf(ax1, g[2]) - fmaxf(ax0, g[0]), 0.0f);
                float ih = fmaxf(fminf(ay1, g[3]) - fmaxf(ay0, g[1]), 0.0f);
                float inter = iw * ih;
                float ua = fmaxf(aarea + barea - inter, 1e-8f);
                iou = inter / ua;
                if (iou >= 0.1f) {
                    // pack (iou, ~anchor) so atomicMax reproduces first-index argmax
                    unsigned long long key =
                        ((unsigned long long)__float_as_uint(iou) << 32) |
                        (unsigned long long)(0xFFFFFFFFu - (unsigned)a);
                    atomicMax(&stag[m], key);
                }
            } else {
                iou = -1.0f;
            }
            if (iou > best) { best = iou; bm = m; }   // strict > => first max index
        }
        iou_max_out[(size_t)b * A + a] = best;
        iou_arg_out[(size_t)b * A + a] = bm;
    }
    __syncthreads();
    for (int m = tid; m < M; m += blockDim.x)
        if (stag[m] != 0ull) atomicMax(&tag_best[(size_t)b * M + m], stag[m]);
}

// ---------------- K2: mark the tag-argmax anchors ----------------
__global__ void posmark_kernel(const unsigned long long* __restrict__ tag_best,
                               const float* __restrict__ annot,
                               uint8_t* __restrict__ pos_tag,
                               int B, int A, int M) {
    int i = blockIdx.x * blockDim.x + threadIdx.x;
    if (i >= B * M) return;
    int b = i / M, m = i - b * M;
    unsigned long long key = tag_best[i];
    if (key == 0ull) return;                                     // no candidate >= 0.1
    if (annot[((size_t)b * M + m) * 5 + 4] == -1.0f) return;     // invalid gt
    unsigned a = 0xFFFFFFFFu - (unsigned)(key & 0xFFFFFFFFu);
    pos_tag[(size_t)b * A + a] = 1;                               // benign same-value race
}

// ---------------- K3: focal / smooth-L1 / femb NLL + WMMA block reduction ----------------
__global__ void __launch_bounds__(256)
loss_kernel(const float* __restrict__ cls_in,   // [B,A,C]
            const float* __restrict__ reg_in,   // [B,A,4]
            const float* __restrict__ anchors,  // [A,4]
            const float* __restrict__ annot,    // [B,M,5]
            const float* __restrict__ stdv,     // [4]
            const float* __restrict__ femb,     // [B,A,F]
            const float* __restrict__ iou_max_in,
            const int*   __restrict__ iou_arg_in,
            const uint8_t* __restrict__ pos_tag,
            float* __restrict__ partials,       // [4][B*nblk]
            int A, int M, int C, int F, int nblk) {
    const int b   = blockIdx.y;
    const int a   = blockIdx.x * blockDim.x + threadIdx.x;
    const int tid = threadIdx.x;

    float cls_s = 0.0f, np = 0.0f, reg_s = 0.0f, femb_s = 0.0f;

    if (a < A) {
        const size_t ba = (size_t)b * A + a;
        const bool pos = (iou_max_in[ba] >= 0.5f) || (pos_tag[ba] != 0);

        // focal classification loss (all anchors)
        const float* cp = cls_in + ba * C;
        for (int c = 0; c < C; ++c) {
            float p = fminf(fmaxf(cp[c], 1e-3f), 1.0f - 1e-3f);
            bool t1 = pos && (c == 0);
            float af  = t1 ? ALPHA_C : (1.0f - ALPHA_C);
            float pt  = t1 ? (1.0f - p) : p;
            float bce = t1 ? -logf(p) : -logf(1.0f - p);
            cls_s += af * pt * pt * bce;
        }

        if (pos) {
            np = 1.0f;
            // ----- smooth-L1 regression vs assigned GT -----
            const int g = iou_arg_in[ba];
            const float* gt = annot + ((size_t)b * M + g) * 5;
            const float ax0 = anchors[a * 4 + 0], ay0 = anchors[a * 4 + 1];
            const float ax1 = anchors[a * 4 + 2], ay1 = anchors[a * 4 + 3];
            const float aw = ax1 - ax0, ah = ay1 - ay0;
            const float acx = ax0 + 0.5f * aw, acy = ay0 + 0.5f * ah;
            float gw = gt[2] - gt[0], gh = gt[3] - gt[1];
            const float gcx = gt[0] + 0.5f * gw, gcy = gt[1] + 0.5f * gh;
            gw = fmaxf(gw, 1.0f); gh = fmaxf(gh, 1.0f);
            float tv[4];
            tv[0] = ((gcx - acx) / aw) / stdv[0];
            tv[1] = ((gcy - acy) / ah) / stdv[1];
            tv[2] = logf(gw / aw) / stdv[2];
            tv[3] = logf(gh / ah) / stdv[3];
            const float* rg = reg_in + ba * 4;
            for (int k = 0; k < 4; ++k) {
                float d = fabsf(tv[k] - rg[k]);
                reg_s += (d <= (1.0f / 9.0f)) ? 4.5f * d * d : (d - 0.5f / 9.0f);
            }
            // ----- femb cross-entropy: gather ONLY positive rows (bandwidth win) -----
            const int label = (int)gt[4];
            const float* fr = femb + ba * F;
            __builtin_prefetch(fr, 0, 0);                 // global_prefetch_b8
            __builtin_prefetch(fr + 64, 0, 0);
            float mx = -INFINITY;
            for (int k = 0; k < F; ++k) mx = fmaxf(mx, fr[k]);
            float se = 0.0f;
            for (int k = 0; k < F; ++k) se += expf(fr[k] - mx);
            femb_s = (mx + logf(se)) - fr[label];
        }
    }

    // ---- block reduction of {cls_s, np, reg_s, femb_s} via fp32 WMMA ones-matmul ----
    __shared__ float red[4 * 256];
    red[0 * 256 + tid] = cls_s;
    red[1 * 256 + tid] = np;
    red[2 * 256 + tid] = reg_s;
    red[3 * 256 + tid] = femb_s;
    __syncthreads();

    if (tid < 32) {   // wave 0, EXEC all ones within the wave (wave32)
        const v2f ones = {1.0f, 1.0f};
        const int blk = b * nblk + blockIdx.x;
        const int per = (int)gridDim.y * nblk;
        for (int q = 0; q < 4; ++q) {
            const float* src = &red[q * 256];
            v8f c = {};
            for (int k = 0; k < 4; ++k) {
                v2f av;
                av.x = src[k * 64 + 2 * tid];
                av.y = src[k * 64 + 2 * tid + 1];
                // D = A(16x4) * ones(4x16) + C : row-sums, fp32-exact adds
                c = __builtin_amdgcn_wmma_f32_16x16x4_f32(
                        false, av, false, ones, (short)0, c, false, false);
            }
            // column N=0 lives in lane 0 (M=0..7) and lane 16 (M=8..15)
            float s = c[0] + c[1] + c[2] + c[3] + c[4] + c[5] + c[6] + c[7];
            s += __shfl_down(s, 16, 32);
            if (tid == 0) partials[(size_t)q * per + blk] = s;
        }
    }
}

// ---------------- K4: deterministic finalize ----------------
__global__ void finalize_kernel(const float* __restrict__ partials,
                                float* __restrict__ out, int B, int nblk) {
    if (threadIdx.x != 0 || blockIdx.x != 0) return;
    const int per = B * nblk;
    float clsm = 0.0f, regm = 0.0f, fembm = 0.0f;
    for (int b = 0; b < B; ++b) {
        float cs = 0.0f, ns = 0.0f, rs = 0.0f, fs = 0.0f;
        for (int i = 0; i < nblk; ++i) {
            cs += partials[0 * per + b * nblk + i];
            ns += partials[1 * per + b * nblk + i];
            rs += partials[2 * per + b * nblk + i];
            fs += partials[3 * per + b * nblk + i];
        }
        clsm  += cs / fmaxf(ns, 1.0f);
        regm  += (ns > 0.0f) ? rs / fmaxf(ns * 4.0f, 1.0f) : 0.0f;
        fembm += (ns > 0.0f) ? fs / fmaxf(ns, 1.0f) : 0.0f;
    }
    out[0] = clsm / (float)B;
    out[1] = regm / (float)B;
    out[2] = fembm / (float)B;
}

// ---------------- launch ----------------
extern "C" void kernel_launch(void* const* d_in, const int* in_sizes, int n_in,
                              void* d_out, int out_size, void* d_ws, size_t ws_size,
                              hipStream_t stream) {
    const float* cls     = (const float*)d_in[0];
    const float* reg     = (const float*)d_in[1];
    const float* anchors = (const float*)d_in[2];  // [1,A,4]
    const float* annot   = (const float*)d_in[3];  // [B,M,5]
    const float* stdv    = (const float*)d_in[4];
    const float* femb    = (const float*)d_in[5];  // [B,A,F]

    const int A = in_sizes[2] / 4;
    const int B = in_sizes[1] / (A * 4);
    const int M = in_sizes[3] / (B * 5);
    const int C = in_sizes[0] / (B * A);
    const int F = in_sizes[5] / (B * A);
    const int nblk = (A + 255) / 256;

    // workspace carve-out
    char* ws = (char*)d_ws;
    float* iou_max_buf = (float*)ws;  ws += (size_t)B * A * sizeof(float);
    int*   iou_arg_buf = (int*)ws;    ws += (size_t)B * A * sizeof(int);
    uint8_t* pos_tag   = (uint8_t*)ws;
    ws += (((size_t)B * A + 255) & ~(size_t)255);          // keep 8B alignment after bytes
    unsigned long long* tag_best = (unsigned long long*)ws;
    ws += (size_t)B * M * sizeof(unsigned long long);
    float* partials = (float*)ws;                           // [4][B*nblk]

    const int nz = (B * A > B * M) ? B * A : B * M;
    zero_ws_kernel<<<(nz + 255) / 256, 256, 0, stream>>>(pos_tag, tag_best, B * A, B * M);

    dim3 grid(nblk, B);
    iou_kernel<<<grid, 256, 0, stream>>>(anchors, annot, iou_max_buf, iou_arg_buf,
                                         tag_best, A, M);
    posmark_kernel<<<(B * M + 255) / 256, 256, 0, stream>>>(tag_best, annot, pos_tag,
                                                            B, A, M);
    loss_kernel<<<grid, 256, 0, stream>>>(cls, reg, anchors, annot, stdv, femb,
                                          iou_max_buf, iou_arg_buf, pos_tag,
                                          partials, A, M, C, F, nblk);
    finalize_kernel<<<1, 32, 0, stream>>>(partials, (float*)d_out, B, nblk);
}